// Conv3d_8529805050235
// MI455X (gfx1250) — compile-verified
//
#include <hip/hip_runtime.h>

// Sparse 3x3x3 conv as gather + tiled GEMM using CDNA5 fp32 WMMA (16x16x4).
//
// out[m, co] = sum_k sum_ci mask[k,m] * feats[in_idx[k,m], ci] * kernel[k, ci, co]
//
// Per wave: one 16-voxel M-tile x full C_OUT=64 (4 N-tiles of 16).
// Contraction length = K_VOL * C_IN = 864 -> 216 WMMA K-steps of 4 -> 864 WMMAs.
// A fragments: direct global b64 gathers per the 16x4 f32 VGPR layout.
// B fragments: kernel staged per-offset into LDS in swizzled fragment order so
// each fragment is a single conflict-free ds_load_b64 (double-buffered 8KB chunks).

typedef __attribute__((ext_vector_type(2))) float v2f;
typedef __attribute__((ext_vector_type(8))) float v8f;

#define N_VOXC 200000
#define M_VOXC 100000
#define K_VOLC 27
#define C_INC  32
#define C_OUTC 64
#define NTILES (M_VOXC / 16)        // 6250 (exact)
#define WPB    8                    // waves per block (256 threads)

__global__ __launch_bounds__(256) void sparse_conv3d_wmma_f32(
    const float* __restrict__ feats,   // [N_VOX, 32]
    const float* __restrict__ kern,    // [27, 32, 64]
    const int*   __restrict__ in_idx,  // [27, M_VOX]
    const int*   __restrict__ maskp,   // [27, M_VOX]
    float*       __restrict__ out)     // [M_VOX, 64]
{
    // [buf][s*128 + ntile*32 + lane] -> (B[k0+2hi][n], B[k0+1+2hi][n])
    __shared__ v2f bfrag[2][1024];     // 2 * 8KB

    const int tid  = threadIdx.x;
    const int lane = tid & 31;
    const int wave = tid >> 5;
    const int lo   = lane & 15;
    const int hi   = lane >> 4;
    const int tile = (int)blockIdx.x * WPB + wave;
    const bool active = (tile < NTILES);   // wave-uniform -> EXEC stays all-1s
    const int m0 = tile * 16;

    // Stage kernel chunk for offset k into LDS buffer buf (all 256 threads).
    auto stage = [&](int buf, int k) {
        #pragma unroll
        for (int i = 0; i < 4; ++i) {
            int e   = tid + i * 256;            // 0..1023
            int el  = e & 31;                   // destination lane
            int nt  = (e >> 5) & 3;             // N-tile
            int s   = e >> 7;                   // K-step within offset (0..7)
            int col = nt * 16 + (el & 15);
            int row = k * C_INC + s * 4 + 2 * (el >> 4);
            v2f b;
            b.x = kern[row * C_OUTC + col];
            b.y = kern[(row + 1) * C_OUTC + col];
            bfrag[buf][e] = b;
        }
    };

    v8f acc0 = {}, acc1 = {}, acc2 = {}, acc3 = {};

    stage(0, 0);

    for (int k = 0; k < K_VOLC; ++k) {
        __syncthreads();                       // chunk k staged; prev reads done
        if (k + 1 < K_VOLC) stage((k + 1) & 1, k + 1);

        if (active) {
            const int   idxv = in_idx[k * M_VOXC + m0 + lo];
            const float ms   = (float)maskp[k * M_VOXC + m0 + lo];
            const float* fr  = feats + idxv * C_INC + 2 * hi;
            const v2f*   bb  = &bfrag[k & 1][lane];
            #pragma unroll
            for (int s = 0; s < 8; ++s) {
                v2f a = *(const v2f*)(fr + s * 4);   // 8B-aligned global b64 gather
                a = a * ms;                          // fold occupancy mask
                acc0 = __builtin_amdgcn_wmma_f32_16x16x4_f32(
                    false, a, false, bb[s * 128 +  0], (short)0, acc0, false, false);
                acc1 = __builtin_amdgcn_wmma_f32_16x16x4_f32(
                    false, a, false, bb[s * 128 + 32], (short)0, acc1, false, false);
                acc2 = __builtin_amdgcn_wmma_f32_16x16x4_f32(
                    false, a, false, bb[s * 128 + 64], (short)0, acc2, false, false);
                acc3 = __builtin_amdgcn_wmma_f32_16x16x4_f32(
                    false, a, false, bb[s * 128 + 96], (short)0, acc3, false, false);
            }
        }
    }

    if (active) {
        // C/D layout: VGPR r -> row m0 + r + 8*hi, col = ntile*16 + lo
        float* orow = out + (m0 + 8 * hi) * C_OUTC + lo;
        #pragma unroll
        for (int r = 0; r < 8; ++r) {
            orow[r * C_OUTC +  0] = acc0[r];
            orow[r * C_OUTC + 16] = acc1[r];
            orow[r * C_OUTC + 32] = acc2[r];
            orow[r * C_OUTC + 48] = acc3[r];
        }
    }
}

extern "C" void kernel_launch(void* const* d_in, const int* in_sizes, int n_in,
                              void* d_out, int out_size, void* d_ws, size_t ws_size,
                              hipStream_t stream) {
    (void)in_sizes; (void)n_in; (void)out_size; (void)d_ws; (void)ws_size;
    const float* feats  = (const float*)d_in[0];
    const float* kern   = (const float*)d_in[1];
    const int*   in_idx = (const int*)d_in[2];
    const int*   maskp  = (const int*)d_in[3];
    float*       out    = (float*)d_out;

    dim3 grid((NTILES + WPB - 1) / WPB);   // 782 blocks
    sparse_conv3d_wmma_f32<<<grid, 256, 0, stream>>>(feats, kern, in_idx, maskp, out);
}